// GraphHeadAttention_19112604467754
// MI455X (gfx1250) — compile-verified
//
#include <hip/hip_runtime.h>
#include <hip/hip_bf16.h>

typedef float v2f __attribute__((ext_vector_type(2)));
typedef float v8f __attribute__((ext_vector_type(8)));

#define Hn 6
#define Nn 1024
#define Cn 384
#define HD 64
#define Mn 4096           // B*N
#define BH 24             // B*H
#define ROWS 24576        // B*H*N

#define WMMA_F32(a, b, c) \
    __builtin_amdgcn_wmma_f32_16x16x4_f32(false, (a), false, (b), (short)0, (c), false, false)

// ---------------------------------------------------------------------------
// fp32 WMMA GEMM:  C[M,N] = op( A[M,K] @ W[N,K]^T + bias )
// One wave computes a 16x64 strip: one A fragment feeds 4 column tiles per
// k-step (4x A reuse).  Single-stage software pipeline: next k-step's 5
// fragments are loaded before the current k-step's 4 wmmas so the scheduler
// can stagger s_wait_loadcnt instead of draining per-wmma.
// Requires M%16==0, N%64==0, K%4==0, K>=8, total tasks % 8 == 0.
// ---------------------------------------------------------------------------
__global__ __launch_bounds__(256) void gemm_wmma_nt(
    const float* __restrict__ A, const float* __restrict__ W,
    const float* __restrict__ bias, float* __restrict__ C,
    int M, int N, int K, int doRelu)
{
    const int lane = threadIdx.x & 31;
    const int wave = threadIdx.x >> 5;
    const int groupsN = N >> 6;                    // 64-wide column groups
    const int task = blockIdx.x * 8 + wave;
    const int tm = task / groupsN;
    const int tg = task - tm * groupsN;
    const int r  = lane & 15;
    const int hi = lane >> 4;                      // 0 or 1
    const int arow = tm * 16 + r;
    const int col0 = tg * 64 + r;

    const float* ap  = A + (size_t)arow * K + 2 * hi;
    const float* wp0 = W + (size_t)col0 * K + 2 * hi;
    const float* wp1 = wp0 + (size_t)16 * K;
    const float* wp2 = wp0 + (size_t)32 * K;
    const float* wp3 = wp0 + (size_t)48 * K;

    __builtin_prefetch(ap, 0, 0);
    __builtin_prefetch(wp0, 0, 0);
    __builtin_prefetch(wp1, 0, 0);
    __builtin_prefetch(wp2, 0, 0);
    __builtin_prefetch(wp3, 0, 0);

    v8f acc0 = {0.f,0.f,0.f,0.f,0.f,0.f,0.f,0.f};
    v8f acc1 = acc0, acc2 = acc0, acc3 = acc0;

    // pipeline prologue
    v2f a_c  = *(const v2f*)(ap);
    v2f b0_c = *(const v2f*)(wp0);
    v2f b1_c = *(const v2f*)(wp1);
    v2f b2_c = *(const v2f*)(wp2);
    v2f b3_c = *(const v2f*)(wp3);

    for (int k0 = 0; k0 < K - 4; k0 += 4) {
        const int kn = k0 + 4;
        // issue next-stage loads BEFORE consuming current fragments
        const v2f a_n  = *(const v2f*)(ap  + kn);
        const v2f b0_n = *(const v2f*)(wp0 + kn);
        const v2f b1_n = *(const v2f*)(wp1 + kn);
        const v2f b2_n = *(const v2f*)(wp2 + kn);
        const v2f b3_n = *(const v2f*)(wp3 + kn);

        acc0 = WMMA_F32(a_c, b0_c, acc0);
        acc1 = WMMA_F32(a_c, b1_c, acc1);
        acc2 = WMMA_F32(a_c, b2_c, acc2);
        acc3 = WMMA_F32(a_c, b3_c, acc3);

        a_c = a_n; b0_c = b0_n; b1_c = b1_n; b2_c = b2_n; b3_c = b3_n;
    }
    // epilogue
    acc0 = WMMA_F32(a_c, b0_c, acc0);
    acc1 = WMMA_F32(a_c, b1_c, acc1);
    acc2 = WMMA_F32(a_c, b2_c, acc2);
    acc3 = WMMA_F32(a_c, b3_c, acc3);

    v8f accs[4] = {acc0, acc1, acc2, acc3};
#pragma unroll
    for (int tt = 0; tt < 4; ++tt) {
        const int col = col0 + tt * 16;
        const float bv = bias ? bias[col] : 0.0f;
#pragma unroll
        for (int q = 0; q < 8; ++q) {
            const int m = tm * 16 + q + 8 * hi;
            float val = accs[tt][q] + bv;
            if (doRelu) val = fmaxf(val, 0.0f);
            C[(size_t)m * N + col] = val;
        }
    }
}

// ---------------------------------------------------------------------------
// si/sj: folds v@Wc^T + weird reshape + dot with a1/a2.
// flat f over (H*N): source (h=f/N, n=f%N), dest (h'=f%H, n'=f/H)
// ---------------------------------------------------------------------------
__global__ __launch_bounds__(256) void sisj_kernel(
    const float* __restrict__ vlin, const float* __restrict__ Wc,
    const float* __restrict__ a, float* __restrict__ si, float* __restrict__ sj)
{
    __shared__ float wc[HD * HD];
    for (int idx = threadIdx.x; idx < HD * HD; idx += 256) wc[idx] = Wc[idx];
    __syncthreads();

    const int f0 = blockIdx.x * 256 + threadIdx.x;   // over B*H*N
    const int b  = f0 / (Hn * Nn);
    const int f  = f0 - b * (Hn * Nn);
    const int h  = f >> 10;
    const int n  = f & 1023;
    const int hp = f % Hn;
    const int np = f / Hn;

    float vr[HD];
    const float* vp = vlin + ((size_t)b * Nn + n) * Cn + h * HD;
#pragma unroll
    for (int e = 0; e < HD; ++e) vr[e] = vp[e];

    const float* a1 = a + hp * (2 * HD);
    const float* a2 = a1 + HD;
    float s1 = 0.f, s2 = 0.f;
    for (int d = 0; d < HD; ++d) {
        const float* wr = wc + d * HD;
        float acc = 0.f;
#pragma unroll 8
        for (int e = 0; e < HD; ++e) acc += vr[e] * wr[e];
        s1 += acc * a1[d];
        s2 += acc * a2[d];
    }
    const size_t o = ((size_t)b * Hn + hp) * Nn + np;
    si[o] = s1; sj[o] = s2;
}

// ---------------------------------------------------------------------------
// Attention scores (WMMA fp32) + softmax stats + top-8 per row.
// One block = one (b,h) x 16-row strip.  q fragment register-resident and
// reused across all 8 column tiles per wave; k-fragment stream is software
// pipelined one stage ahead.  Scores tile 16x1024 in LDS (64KB).
// Row reductions use 16-lane shuffle groups (each row group lies in one wave).
// ---------------------------------------------------------------------------
__global__ __launch_bounds__(256) void attn_score_topk(
    const float* __restrict__ qlin, const float* __restrict__ klin,
    int* __restrict__ topk_idx)
{
    __shared__ float sc[16 * Nn];   // 64 KB
    const int bid  = blockIdx.x;
    const int iblk = bid & 63;      // N/16 = 64 strips
    const int bh   = bid >> 6;
    const int b    = bh / Hn, h = bh - b * Hn;
    const int lane = threadIdx.x & 31, wave = threadIdx.x >> 5;
    const int r    = lane & 15, hi = lane >> 4;
    const int i0   = iblk * 16;
    const float scale = 0.125f;     // hd^-0.5

    const size_t base = (size_t)b * Nn * Cn + h * HD;
    const float* qp = qlin + base + (size_t)(i0 + r) * Cn + 2 * hi;

    // hoist q fragment (K=64 -> 16 k-steps) into registers: reused 8x
    v2f qa[16];
#pragma unroll
    for (int ks = 0; ks < 16; ++ks) qa[ks] = *(const v2f*)(qp + 4 * ks);

    for (int jt = wave; jt < 64; jt += 8) {
        const float* kp = klin + base + (size_t)(jt * 16 + r) * Cn + 2 * hi;
        v8f acc = {0.f,0.f,0.f,0.f,0.f,0.f,0.f,0.f};
        v2f b_c = *(const v2f*)(kp);
#pragma unroll
        for (int ks = 0; ks < 15; ++ks) {
            const v2f b_n = *(const v2f*)(kp + 4 * (ks + 1));  // next stage first
            acc = WMMA_F32(qa[ks], b_c, acc);
            b_c = b_n;
        }
        acc = WMMA_F32(qa[15], b_c, acc);
#pragma unroll
        for (int q = 0; q < 8; ++q)
            sc[(q + 8 * hi) * Nn + jt * 16 + r] = acc[q] * scale;
    }
    __syncthreads();

    // per-row phase: 16 threads per row, contiguous inside one wave
    const int row = threadIdx.x >> 4;
    const int t   = threadIdx.x & 15;
    float* srow = sc + row * Nn;

    float mx = -__builtin_inff();
    for (int u = 0; u < 64; ++u) mx = fmaxf(mx, srow[t + 16 * u]);
    for (int off = 8; off; off >>= 1) mx = fmaxf(mx, __shfl_xor(mx, off, 16));

    float sm = 0.f;
    for (int u = 0; u < 64; ++u) sm += expf(srow[t + 16 * u] - mx);
    for (int off = 8; off; off >>= 1) sm += __shfl_xor(sm, off, 16);

    int* orow = topk_idx + ((size_t)bh * Nn + i0 + row) * 8;
    for (int p = 0; p < 8; ++p) {
        float bv = -__builtin_inff(); int bj = Nn;
        for (int u = 0; u < 64; ++u) {
            const int j = t + 16 * u;
            const float v = srow[j];
            if (v > bv || (v == bv && j < bj)) { bv = v; bj = j; }
        }
        for (int off = 8; off; off >>= 1) {
            const float ov = __shfl_xor(bv, off, 16);
            const int   oj = __shfl_xor(bj, off, 16);
            if (ov > bv || (ov == bv && oj < bj)) { bv = ov; bj = oj; }
        }
        if (t == 0) {
            const float prob = expf(bv - mx) / sm;   // == softmax value
            orow[p] = (prob > 0.f) ? bj : -1;
        }
        if (t == (bj & 15)) srow[bj] = -__builtin_inff();  // intra-wave, lockstep
    }
}

// ---------------------------------------------------------------------------
__global__ void zero_u32(unsigned* __restrict__ p, int n)
{
    const int i = blockIdx.x * 256 + threadIdx.x;
    if (i < n) p[i] = 0u;
}

__global__ void build_mask(const int* __restrict__ topk, unsigned* __restrict__ mask)
{
    const int ri = blockIdx.x * 256 + threadIdx.x;   // over B*H*N
    if (ri >= ROWS) return;
    const int bh = ri >> 10;
    const int i  = ri & 1023;
    unsigned* mrow = mask + (size_t)bh * Nn * 32;
    atomicOr(&mrow[(size_t)i * 32 + (i >> 5)], 1u << (i & 31));   // diagonal
    const int* tk = topk + (size_t)ri * 8;
    for (int p = 0; p < 8; ++p) {
        const int j = tk[p];
        if (j >= 0) {
            atomicOr(&mrow[(size_t)i * 32 + (j >> 5)], 1u << (j & 31));
            atomicOr(&mrow[(size_t)j * 32 + (i >> 5)], 1u << (i & 31));
        }
    }
}

// ---------------------------------------------------------------------------
// Sparse graph softmax + aggregation.  One block (128 thr) per (b,h,i) row.
// ---------------------------------------------------------------------------
__global__ __launch_bounds__(128) void graph_attn(
    const float* __restrict__ si, const float* __restrict__ sj,
    const unsigned* __restrict__ mask, const float* __restrict__ vlin,
    float* __restrict__ out2)
{
    __shared__ float wlist[Nn];
    __shared__ int   jlist[Nn];
    __shared__ int   cnt;
    __shared__ float redmax[4];
    __shared__ float redsum[4];

    const int ri = blockIdx.x;           // bh*N + i
    const int bh = ri >> 10, i = ri & 1023;
    const int b  = bh / Hn, h = bh - b * Hn;
    const int tid = threadIdx.x;
    if (tid == 0) cnt = 0;
    __syncthreads();

    const unsigned* mrow = mask + (size_t)ri * 32;
    const float siv = si[(size_t)bh * Nn + i];
    const float* sjrow = sj + (size_t)bh * Nn;

    // pass 1: row max of leaky_relu(si+sj) over masked cols
    float mx = -__builtin_inff();
    for (int j = tid; j < Nn; j += 128) {
        if ((mrow[j >> 5] >> (j & 31)) & 1u) {
            float e = siv + sjrow[j];
            e = (e > 0.f) ? e : 0.2f * e;
            mx = fmaxf(mx, e);
        }
    }
    for (int off = 16; off; off >>= 1) mx = fmaxf(mx, __shfl_xor(mx, off, 32));
    if ((tid & 31) == 0) redmax[tid >> 5] = mx;
    __syncthreads();
    mx = fmaxf(fmaxf(redmax[0], redmax[1]), fmaxf(redmax[2], redmax[3]));
    __syncthreads();

    // pass 2: exp, sum, compact neighbor list
    float sm = 0.f;
    for (int j = tid; j < Nn; j += 128) {
        if ((mrow[j >> 5] >> (j & 31)) & 1u) {
            float e = siv + sjrow[j];
            e = (e > 0.f) ? e : 0.2f * e;
            const float w = expf(e - mx);
            sm += w;
            const int slot = atomicAdd(&cnt, 1);
            wlist[slot] = w; jlist[slot] = j;
        }
    }
    for (int off = 16; off; off >>= 1) sm += __shfl_xor(sm, off, 32);
    if ((tid & 31) == 0) redsum[tid >> 5] = sm;
    __syncthreads();
    sm = redsum[0] + redsum[1] + redsum[2] + redsum[3];
    const int n = cnt;

    if (tid < HD) {
        float acc = 0.f;
        const float* vb = vlin + (size_t)b * Nn * Cn + h * HD + tid;
        for (int t = 0; t < n; ++t)
            acc += wlist[t] * vb[(size_t)jlist[t] * Cn];
        out2[((size_t)b * Nn + i) * Cn + h * HD + tid] = acc / sm;
    }
}

// ---------------------------------------------------------------------------
extern "C" void kernel_launch(void* const* d_in, const int* in_sizes, int n_in,
                              void* d_out, int out_size, void* d_ws, size_t ws_size,
                              hipStream_t stream)
{
    const float* x    = (const float*)d_in[0];
    const float* Wq   = (const float*)d_in[1];
    const float* bq   = (const float*)d_in[2];
    const float* Wk   = (const float*)d_in[3];
    const float* bk   = (const float*)d_in[4];
    const float* Wv   = (const float*)d_in[5];
    const float* bv   = (const float*)d_in[6];
    const float* Wc   = (const float*)d_in[7];
    const float* av   = (const float*)d_in[8];
    const float* Wgat = (const float*)d_in[9];
    const float* Wp   = (const float*)d_in[10];
    const float* bp   = (const float*)d_in[11];

    const size_t MC = (size_t)Mn * Cn;          // 1,572,864
    float* q    = (float*)d_ws;
    float* k    = q + MC;
    float* v    = k + MC;
    float* out2 = v + MC;
    float* g    = out2 + MC;
    float* si   = g + MC;
    float* sj   = si + ROWS;
    int*      topk = (int*)(sj + ROWS);
    unsigned* mask = (unsigned*)(topk + ROWS * 8);

    // tasks = (M/16)*(N/64) = 256*6 = 1536 ; 8 waves/block -> 192 blocks
    const int gemmBlocks = (Mn / 16) * (Cn / 64) / 8;

    // QKV projections (fp32 WMMA)
    gemm_wmma_nt<<<gemmBlocks, 256, 0, stream>>>(x, Wq, bq, q, Mn, Cn, Cn, 0);
    gemm_wmma_nt<<<gemmBlocks, 256, 0, stream>>>(x, Wk, bk, k, Mn, Cn, Cn, 0);
    gemm_wmma_nt<<<gemmBlocks, 256, 0, stream>>>(x, Wv, bv, v, Mn, Cn, Cn, 0);

    // graph scores si/sj (folds v@Wc^T + reshape permutation)
    sisj_kernel<<<ROWS / 256, 256, 0, stream>>>(v, Wc, av, si, sj);

    // q.k^T + softmax stats + top-8 (fp32 WMMA)
    attn_score_topk<<<BH * (Nn / 16), 256, 0, stream>>>(q, k, topk);

    // adjacency bitmask
    zero_u32<<<(ROWS * 32) / 256, 256, 0, stream>>>(mask, ROWS * 32);
    build_mask<<<ROWS / 256, 256, 0, stream>>>(topk, mask);

    // sparse graph softmax + aggregation -> (B,N,C)
    graph_attn<<<ROWS, 128, 0, stream>>>(si, sj, mask, v, out2);

    // output projections (fp32 WMMA)
    gemm_wmma_nt<<<gemmBlocks, 256, 0, stream>>>(out2, Wgat, nullptr, g, Mn, Cn, Cn, 1);
    gemm_wmma_nt<<<gemmBlocks, 256, 0, stream>>>(g, Wp, bp, (float*)d_out, Mn, Cn, Cn, 0);
}